// TinyGATLayer_49409303773457
// MI455X (gfx1250) — compile-verified
//
#include <hip/hip_runtime.h>
#include <hip/hip_bf16.h>
#include <math.h>

typedef __attribute__((ext_vector_type(2))) float v2f;
typedef __attribute__((ext_vector_type(8))) float v8f;

#define BATCH 8
#define NROW  2048
#define DIN   256
#define DOUT  128
#define TOPK  16

// ---------------------------------------------------------------------------
// Kernel 1: w2[i] = sum_o W[o,i] * a2[o]   (a2 = a[DOUT..2*DOUT))
// 1 block x 256 threads; per-o accesses are coalesced across threads.
// ---------------------------------------------------------------------------
__global__ void k_w2(const float* __restrict__ W, const float* __restrict__ a,
                     float* __restrict__ w2) {
    int i = threadIdx.x;                     // 0..255
    const float* a2 = a + DOUT;
    float s = 0.f;
#pragma unroll 4
    for (int o = 0; o < DOUT; ++o) s = fmaf(W[o * DIN + i], a2[o], s);
    w2[i] = s;
}

// ---------------------------------------------------------------------------
// Kernel 2: sj[r] = dot(x[r, :], w2)  for r in [0, BATCH*NROW)
// One wave32 per row, float4 loads, shuffle tree reduction.
// ---------------------------------------------------------------------------
__global__ void k_sj(const float* __restrict__ x, const float* __restrict__ w2,
                     float* __restrict__ sj) {
    __shared__ float lw2[DIN];
    int tid = threadIdx.x;                   // 256 threads = 8 waves
    lw2[tid] = w2[tid];
    __syncthreads();

    int wave = tid >> 5;
    int lane = tid & 31;
    int row  = blockIdx.x * 8 + wave;        // 2048 blocks * 8 = 16384 rows
    const float4* xr = (const float4*)(x + (size_t)row * DIN);
    const float4* w4 = (const float4*)lw2;

    float4 p0 = xr[lane * 2 + 0], p1 = xr[lane * 2 + 1];
    float4 q0 = w4[lane * 2 + 0], q1 = w4[lane * 2 + 1];
    float s = p0.x * q0.x + p0.y * q0.y + p0.z * q0.z + p0.w * q0.w
            + p1.x * q1.x + p1.y * q1.y + p1.z * q1.z + p1.w * q1.w;

#pragma unroll
    for (int off = 16; off >= 1; off >>= 1) s += __shfl_xor(s, off, 32);
    if (lane == 0) sj[row] = s;
}

// ---------------------------------------------------------------------------
// Kernel 3: per batch, top-16 of sj[b,:] (stable: ties -> smallest index,
// values selected in descending order like lax.top_k) + softmax weights.
// One block per batch.
// ---------------------------------------------------------------------------
__global__ void k_topk(const float* __restrict__ sj, int* __restrict__ topi,
                       float* __restrict__ attn) {
    __shared__ float vals[NROW];
    __shared__ float rv[256];
    __shared__ int   ri[256];
    __shared__ float tv[TOPK];

    int b = blockIdx.x, tid = threadIdx.x;
    const float* rowp = sj + b * NROW;
#pragma unroll
    for (int t = 0; t < NROW / 256; ++t) vals[tid + t * 256] = rowp[tid + t * 256];
    __syncthreads();

    const float NEG = -__builtin_inff();
    for (int it = 0; it < TOPK; ++it) {
        // contiguous chunk per thread; strict '>' keeps lowest index on ties
        int base = tid * (NROW / 256);
        float bv = NEG; int bi = base;
#pragma unroll
        for (int t = 0; t < NROW / 256; ++t) {
            float v = vals[base + t];
            if (v > bv) { bv = v; bi = base + t; }
        }
        rv[tid] = bv; ri[tid] = bi;
        __syncthreads();
        for (int s = 128; s > 0; s >>= 1) {
            if (tid < s) {
                float ov = rv[tid + s]; int oi = ri[tid + s];
                if (ov > rv[tid] || (ov == rv[tid] && oi < ri[tid])) {
                    rv[tid] = ov; ri[tid] = oi;
                }
            }
            __syncthreads();
        }
        if (tid == 0) {
            tv[it] = rv[0];
            topi[b * TOPK + it] = ri[0];
            vals[ri[0]] = NEG;               // remove for next round
        }
        __syncthreads();
    }

    // softmax over the 16 selected values (tv[0] is the max: descending order).
    // The si[b,n] offset of the reference cancels exactly here.
    if (tid < TOPK) {
        float m = tv[0];
        float sum = 0.f;
#pragma unroll
        for (int k = 0; k < TOPK; ++k) sum += expf(tv[k] - m);
        attn[b * TOPK + tid] = expf(tv[tid] - m) / sum;
    }
}

// ---------------------------------------------------------------------------
// Kernel 4: per (batch, 16-col tile) wave:
//   Htop(16x16 tile) = Xgather(16x256) @ W^T(256x16tile) via V_WMMA_F32_16X16X4_F32
//   then outvec[b, nt*16+n] = sum_m attn[b,m] * Htop[m,n] folded on the
//   accumulator VGPRs (+ one cross-half shuffle).
// grid = (BATCH, DOUT/16), block = 32 (one wave, EXEC all ones for WMMA).
// ---------------------------------------------------------------------------
__global__ void k_wmma_out(const float* __restrict__ x, const float* __restrict__ W,
                           const int* __restrict__ topi, const float* __restrict__ attn,
                           float* __restrict__ outvec) {
    int b    = blockIdx.x;        // 0..7
    int nt   = blockIdx.y;        // 0..7
    int lane = threadIdx.x;       // 0..31
    int m    = lane & 15;         // M (A) / N (B,D) index for this lane
    int half = lane >> 4;         // 0: K={0,1} of each step, 1: K={2,3}

    int row = topi[b * TOPK + m];                               // gathered x row
    const float* xr = x + ((size_t)b * NROW + row) * DIN + 2 * half;  // A: M=m
    const float* wr = W + (size_t)(nt * 16 + m) * DIN + 2 * half;     // B: N=m (B=W^T)

    v8f acc = {};
#pragma unroll 8
    for (int k0 = 0; k0 < DIN; k0 += 4) {
        v2f av; av.x = xr[k0]; av.y = xr[k0 + 1];
        v2f bv; bv.x = wr[k0]; bv.y = wr[k0 + 1];
        acc = __builtin_amdgcn_wmma_f32_16x16x4_f32(
            /*neg_a=*/false, av, /*neg_b=*/false, bv,
            /*c_mod=*/(short)0, acc, /*reuse_a=*/false, /*reuse_b=*/false);
    }

    // D layout: lane holds N = lane%16; VGPR v holds M = 8*half + v.
    float part = 0.f;
#pragma unroll
    for (int v = 0; v < 8; ++v)
        part = fmaf(attn[b * TOPK + half * 8 + v], acc[v], part);
    part += __shfl_down(part, 16, 32);       // combine M=0..7 with M=8..15
    if (lane < 16) outvec[b * DOUT + nt * 16 + lane] = part;
}

// ---------------------------------------------------------------------------
// Kernel 5: broadcast outvec[b, :] to out[b, n, :] for all n (float4 stores).
// 8*2048*128 floats = 524288 float4 -> grid 2048 x 256.
// ---------------------------------------------------------------------------
__global__ void k_bcast(const float* __restrict__ outvec, float* __restrict__ out) {
    size_t gid = (size_t)blockIdx.x * blockDim.x + threadIdx.x;  // float4 id
    int    c4  = (int)(gid & 31);            // 32 float4 per 128-float row
    size_t rw  = gid >> 5;                   // row id 0..16383
    int    b   = (int)(rw >> 11);            // /2048
    ((float4*)out)[gid] = ((const float4*)outvec)[b * 32 + c4];
}

// ---------------------------------------------------------------------------
extern "C" void kernel_launch(void* const* d_in, const int* in_sizes, int n_in,
                              void* d_out, int out_size, void* d_ws, size_t ws_size,
                              hipStream_t stream) {
    const float* x = (const float*)d_in[0];   // (8, 2048, 256) f32
    const float* W = (const float*)d_in[1];   // (128, 256) f32
    const float* a = (const float*)d_in[2];   // (256,) f32
    float* out = (float*)d_out;               // (8, 2048, 128) f32

    // Workspace layout (all regions written before read each call).
    char* ws = (char*)d_ws;
    float* w2     = (float*)(ws + 0);                       // 256 f   (1 KB)
    float* sj     = (float*)(ws + 1024);                    // 16384 f (64 KB)
    int*   topi   = (int*)  (ws + 1024 + 65536);            // 128 i   (512 B)
    float* attn   = (float*)(ws + 1024 + 65536 + 512);      // 128 f   (512 B)
    float* outvec = (float*)(ws + 1024 + 65536 + 1024);     // 1024 f  (4 KB)

    k_w2      <<<1, 256, 0, stream>>>(W, a, w2);
    k_sj      <<<(BATCH * NROW) / 8, 256, 0, stream>>>(x, w2, sj);
    k_topk    <<<BATCH, 256, 0, stream>>>(sj, topi, attn);
    k_wmma_out<<<dim3(BATCH, DOUT / 16), 32, 0, stream>>>(x, W, topi, attn, outvec);
    k_bcast   <<<(BATCH * NROW * DOUT / 4) / 256, 256, 0, stream>>>(outvec, out);
}